// Attention_49804440764504
// MI455X (gfx1250) — compile-verified
//
#include <hip/hip_runtime.h>
#include <hip/hip_bf16.h>
#include <math.h>
#include <stdint.h>

// ---------------- problem constants ----------------
constexpr int B    = 8;
constexpr int S    = 4;
constexpr int D    = 2560;
constexpr int H    = 32;
constexpr int KVH  = 8;
constexpr int HD   = 128;
constexpr int PAST = 8192;
constexpr int TTOT = PAST + S;         // 8196
constexpr int REP  = H / KVH;          // 4
constexpr int M    = B * S;            // 32 rows for all projections
constexpr int NQ   = H * HD;           // 4096
constexpr int NKV  = KVH * HD;         // 1024
constexpr int NTOT = NQ + 2 * NKV;     // 6144

constexpr int NCHUNK = 16;
constexpr int CHUNK  = 513;            // 16*513 = 8208 >= 8196

// workspace layout (float offsets)
constexpr size_t O_QRAW = 0;                         // 32*4096
constexpr size_t O_KRAW = O_QRAW + (size_t)M * NQ;   // 32*1024
constexpr size_t O_VRAW = O_KRAW + (size_t)M * NKV;
constexpr size_t O_QROT = O_VRAW + (size_t)M * NKV;  // (B,H,S,HD)
constexpr size_t O_M    = O_QROT + (size_t)B * H * S * HD;   // B*H*S*NCHUNK
constexpr size_t O_L    = O_M + (size_t)B * H * S * NCHUNK;
constexpr size_t O_ACC  = O_L + (size_t)B * H * S * NCHUNK;  // *HD
constexpr size_t O_ATTN = O_ACC + (size_t)B * H * S * NCHUNK * HD; // (B*S, H*HD)

typedef __attribute__((ext_vector_type(2))) float v2f;
typedef __attribute__((ext_vector_type(8))) float v8f;

// One wave computes a 16x16 f32 C tile: C = A(16xK) * Bw(16xK)^T
__device__ inline v8f wmma_gemm_tile(const float* __restrict__ A, int lda,
                                     const float* __restrict__ Bw, int ldw,
                                     int K) {
  const int lane = threadIdx.x & 31;
  const int row  = lane & 15;
  const int koff = (lane >> 4) << 1;   // 0 or 2
  v8f c = {0.f, 0.f, 0.f, 0.f, 0.f, 0.f, 0.f, 0.f};
  for (int k0 = 0; k0 < K; k0 += 4) {
    v2f a = *(const v2f*)(A  + (size_t)row * lda + k0 + koff);
    v2f b = *(const v2f*)(Bw + (size_t)row * ldw + k0 + koff);
    c = __builtin_amdgcn_wmma_f32_16x16x4_f32(false, a, false, b, (short)0, c,
                                              false, false);
  }
  return c;
}

__device__ inline void wmma_store_tile(v8f c, float* __restrict__ dst, int ldd) {
  const int lane = threadIdx.x & 31;
  const int half = lane >> 4;
  const int col  = lane & 15;
#pragma unroll
  for (int j = 0; j < 8; ++j) {
    dst[(size_t)(j + half * 8) * ldd + col] = c[j];
  }
}

// ---------------- kernel 1: fused QKV projection (WMMA f32) ----------------
__global__ void qkv_proj_kernel(const float* __restrict__ x,
                                const float* __restrict__ wq,
                                const float* __restrict__ wk,
                                const float* __restrict__ wv,
                                float* __restrict__ ws) {
  const int wave  = threadIdx.x >> 5;
  const int tile  = blockIdx.x * (blockDim.x >> 5) + wave;
  const int ntn   = NTOT / 16;              // 384
  if (tile >= 2 * ntn) return;
  const int mt = tile / ntn;
  const int ng = (tile % ntn) * 16;

  const float* W; float* dst; int nloc, ldd;
  if (ng < NQ)            { W = wq + (size_t)ng * D;              dst = ws + O_QRAW; nloc = ng;            ldd = NQ;  }
  else if (ng < NQ + NKV) { W = wk + (size_t)(ng - NQ) * D;       dst = ws + O_KRAW; nloc = ng - NQ;       ldd = NKV; }
  else                    { W = wv + (size_t)(ng - NQ - NKV) * D; dst = ws + O_VRAW; nloc = ng - NQ - NKV; ldd = NKV; }

  v8f c = wmma_gemm_tile(x + (size_t)(mt * 16) * D, D, W, D, D);
  wmma_store_tile(c, dst + (size_t)(mt * 16) * ldd + nloc, ldd);
}

// ---------------- kernel 2: RoPE q,k + scatter new k,v into output cache ---
__global__ void rope_scatter_kernel(float* __restrict__ ws,
                                    float* __restrict__ outK,
                                    float* __restrict__ outV) {
  const int idx = blockIdx.x * blockDim.x + threadIdx.x;
  const float LOG_THETA = 15.42494847f;   // ln(5e6)
  if (idx < M * NQ) {                     // ---- q ----
    const int m = idx / NQ, n = idx % NQ;
    const int h = n >> 7, hd = n & 127;
    const int b = m >> 2, s = m & 3;
    const int j = hd & 63;
    const float inv = __expf(-((float)j / 64.0f) * LOG_THETA);
    const float ang = (float)(PAST + s) * inv;
    float sn, cs; __sincosf(ang, &sn, &cs);
    const float* qr = ws + O_QRAW + (size_t)m * NQ + h * HD;
    const float x1 = qr[j], x2 = qr[j + 64];
    const float val = (hd < 64) ? (x1 * cs - x2 * sn) : (x2 * cs + x1 * sn);
    ws[O_QROT + ((size_t)(b * H + h) * S + s) * HD + hd] = val;
  } else if (idx < M * NQ + M * NKV) {    // ---- k ----
    const int i2 = idx - M * NQ;
    const int m = i2 / NKV, n = i2 % NKV;
    const int kv = n >> 7, hd = n & 127;
    const int b = m >> 2, s = m & 3;
    const int j = hd & 63;
    const float inv = __expf(-((float)j / 64.0f) * LOG_THETA);
    const float ang = (float)(PAST + s) * inv;
    float sn, cs; __sincosf(ang, &sn, &cs);
    const float* kr = ws + O_KRAW + (size_t)m * NKV + kv * HD;
    const float x1 = kr[j], x2 = kr[j + 64];
    const float val = (hd < 64) ? (x1 * cs - x2 * sn) : (x2 * cs + x1 * sn);
    outK[((size_t)(b * KVH + kv) * TTOT + PAST + s) * HD + hd] = val;
  } else if (idx < M * NQ + 2 * M * NKV) { // ---- v (plain copy) ----
    const int i2 = idx - M * NQ - M * NKV;
    const int m = i2 / NKV, n = i2 % NKV;
    const int kv = n >> 7, hd = n & 127;
    const int b = m >> 2, s = m & 3;
    outV[((size_t)(b * KVH + kv) * TTOT + PAST + s) * HD + hd] =
        ws[O_VRAW + (size_t)m * NKV + n];
  }
}

// ---------------- kernel 3: WMMA flash attention, fused cache copy ---------
// One wave per (b, kv, chunk). 16 query rows = 4 GQA heads x 4 positions.
// Q tile lives in registers as pre-scaled A-fragments; per 16-key tile:
//   scores = 32x v_wmma_f32_16x16x4_f32 (K fragments double as cache copy)
//   V tile staged LDS via global_load_async_to_lds_b128, copied out with
//   global_store_async_from_lds_b128; PV = 32 more WMMAs into reg acc.
__global__ void __launch_bounds__(32)
flash_wmma_kernel(const float* __restrict__ cacheK,
                  const float* __restrict__ cacheV,
                  float* __restrict__ ws,
                  float* __restrict__ outK,
                  float* __restrict__ outV) {
  __shared__ float vtile[16][HD];    // 8 KB: V tile (async staged)
  __shared__ float sc[16][16];       // scores -> weights
  __shared__ float scbuf[16];        // per-row acc rescale factor

  const int lane = threadIdx.x & 31;
  const int col  = lane & 15;
  const int half = lane >> 4;        // 0/1
  const int koff = half << 1;        // 0/2

  const int chunk = blockIdx.x % NCHUNK;
  const int kvb   = blockIdx.x / NCHUNK;
  const int kv    = kvb % KVH;
  const int b     = kvb / KVH;
  const size_t gkv = (size_t)(b * KVH + kv);

  // ---- Q tile -> A-fragments in registers, pre-scaled by 1/sqrt(HD) ----
  const float scaleS = 0.08838834764831845f;
  const float* qptr = ws + O_QROT +
      ((size_t)(b * H + kv * REP + (col >> 2)) * S + (col & 3)) * HD;
  v2f qf[32];
#pragma unroll
  for (int i = 0; i < 32; ++i) {
    v2f q = *(const v2f*)(qptr + i * 4 + koff);
    qf[i].x = q.x * scaleS;
    qf[i].y = q.y * scaleS;
  }

  v8f accf[8];
#pragma unroll
  for (int n = 0; n < 8; ++n)
    accf[n] = (v8f){0.f, 0.f, 0.f, 0.f, 0.f, 0.f, 0.f, 0.f};

  float mrun = -__builtin_inff();    // meaningful on lanes 0-15 (row = lane)
  float lrun = 0.f;

  const int tstart = chunk * CHUNK;
  const int tend   = (tstart + CHUNK < TTOT) ? tstart + CHUNK : TTOT;

  for (int tb = tstart; tb < tend; tb += 16) {
    const int nvalid = (tend - tb < 16) ? (tend - tb) : 16;
    int trow = tb + col;
    const bool valid = trow < tend;
    if (!valid) trow = tb;                 // clamp to a finite row
    const bool past   = trow < PAST;
    const bool docopy = past && valid;

    // ---- stage V tile into LDS (async), overlapped with QK WMMAs ----
    asm volatile("s_wait_asynccnt 0x0" ::: "memory");  // vtile reusable
    {
      const float* vsrc = past ? (cacheV + (gkv * PAST + trow) * HD)
                               : (outV   + (gkv * TTOT + trow) * HD);
      const float* g = vsrc + half * 64;
      uint32_t lds = (uint32_t)(uintptr_t)&vtile[col][half * 64];
#pragma unroll
      for (int i = 0; i < 16; ++i) {
        asm volatile("global_load_async_to_lds_b128 %0, %1, off"
                     :: "v"(lds + i * 16), "v"(g + i * 4) : "memory");
      }
    }

    // ---- scores = Q @ K^T via WMMA; K fragment loads double as copy ----
    const float* ksrc = past ? (cacheK + (gkv * PAST + trow) * HD)
                             : (outK   + (gkv * TTOT + trow) * HD);
    float* kdst = outK + (gkv * TTOT + trow) * HD;
    v8f s8 = {0.f, 0.f, 0.f, 0.f, 0.f, 0.f, 0.f, 0.f};
#pragma unroll
    for (int i = 0; i < 32; ++i) {
      v2f kf = *(const v2f*)(ksrc + i * 4 + koff);
      if (docopy) *(v2f*)(kdst + i * 4 + koff) = kf;   // fused cache copy
      s8 = __builtin_amdgcn_wmma_f32_16x16x4_f32(false, qf[i], false, kf,
                                                 (short)0, s8, false, false);
    }
#pragma unroll
    for (int j = 0; j < 8; ++j) sc[j + half * 8][col] = s8[j];
    __syncthreads();

    // ---- online softmax: lanes 0-15, one query row each ----
    if (lane < 16) {
      float vals[16];
#pragma unroll
      for (int j = 0; j < 16; ++j) vals[j] = sc[lane][j];
      float mc = -__builtin_inff();
      for (int j = 0; j < 16; ++j)
        if (j < nvalid) mc = fmaxf(mc, vals[j]);
      const float mnew = fmaxf(mrun, mc);
      const float scl  = (mrun == -__builtin_inff()) ? 0.f
                                                     : __expf(mrun - mnew);
      float ssum = 0.f;
#pragma unroll
      for (int j = 0; j < 16; ++j) {
        const float w = (j < nvalid) ? __expf(vals[j] - mnew) : 0.f;
        sc[lane][j] = w;
        ssum += w;
      }
      lrun = lrun * scl + ssum;
      mrun = mnew;
      scbuf[lane] = scl;
    }
    __syncthreads();

    // ---- V tile ready: issue async LDS -> HBM cache copy ----
    asm volatile("s_wait_asynccnt 0x0" ::: "memory");
    if (docopy) {
      float* g = outV + (gkv * TTOT + trow) * HD + half * 64;
      uint32_t lds = (uint32_t)(uintptr_t)&vtile[col][half * 64];
#pragma unroll
      for (int i = 0; i < 16; ++i) {
        asm volatile("global_store_async_from_lds_b128 %0, %1, off"
                     :: "v"(g + i * 4), "v"(lds + i * 16) : "memory");
      }
    }

    // ---- rescale acc fragments ----
    {
      float rs[8];
#pragma unroll
      for (int j = 0; j < 8; ++j) rs[j] = scbuf[j + half * 8];
#pragma unroll
      for (int n = 0; n < 8; ++n)
#pragma unroll
        for (int j = 0; j < 8; ++j) accf[n][j] *= rs[j];
    }

    // ---- acc += W @ V via WMMA (A from sc, B from vtile) ----
    v2f wf[4];
#pragma unroll
    for (int k = 0; k < 4; ++k)
      wf[k] = *(const v2f*)&sc[col][k * 4 + koff];
#pragma unroll
    for (int n = 0; n < 8; ++n) {
#pragma unroll
      for (int k = 0; k < 4; ++k) {
        v2f bf;
        bf.x = vtile[k * 4 + koff][n * 16 + col];
        bf.y = vtile[k * 4 + koff + 1][n * 16 + col];
        accf[n] = __builtin_amdgcn_wmma_f32_16x16x4_f32(
            false, wf[k], false, bf, (short)0, accf[n], false, false);
      }
    }
    __syncthreads();   // sc/vtile consumed before next tile overwrites
  }

  // ---- write split-T partials ----
  asm volatile("s_wait_asynccnt 0x0" ::: "memory");
  if (lane < 16) {
    const int r = lane >> 2, s = lane & 3;
    const size_t pidx =
        ((size_t)(b * H + kv * REP + r) * S + s) * NCHUNK + chunk;
    ws[O_M + pidx] = mrun;
    ws[O_L + pidx] = lrun;
  }
#pragma unroll
  for (int j = 0; j < 8; ++j) {
    const int rowid = j + half * 8;
    const int r = rowid >> 2, s = rowid & 3;
    const size_t pidx =
        ((size_t)(b * H + kv * REP + r) * S + s) * NCHUNK + chunk;
#pragma unroll
    for (int n = 0; n < 8; ++n)
      ws[O_ACC + pidx * HD + n * 16 + col] = accf[n][j];
  }
}

// ---------------- kernel 4: combine split-T partials -----------------------
__global__ void combine_kernel(float* __restrict__ ws) {
  const int row = blockIdx.x;                  // ((b*H+h)*S+s)
  const int d   = threadIdx.x;                 // 0..127
  float mstar = -__builtin_inff();
  for (int c = 0; c < NCHUNK; ++c)
    mstar = fmaxf(mstar, ws[O_M + (size_t)row * NCHUNK + c]);
  float lsum = 0.f, osum = 0.f;
  for (int c = 0; c < NCHUNK; ++c) {
    const float e = __expf(ws[O_M + (size_t)row * NCHUNK + c] - mstar);
    lsum += e * ws[O_L + (size_t)row * NCHUNK + c];
    osum += e * ws[O_ACC + ((size_t)row * NCHUNK + c) * HD + d];
  }
  const int s = row & 3, h = (row >> 2) & 31, b = row >> 7;
  ws[O_ATTN + (size_t)(b * S + s) * NQ + h * HD + d] = osum / lsum;
}

// ---------------- kernel 5: output projection (WMMA f32) -------------------
__global__ void out_proj_kernel(const float* __restrict__ ws,
                                const float* __restrict__ wo,
                                float* __restrict__ out) {
  const int wave = threadIdx.x >> 5;
  const int tile = blockIdx.x * (blockDim.x >> 5) + wave;
  const int ntn  = D / 16;                     // 160
  if (tile >= 2 * ntn) return;
  const int mt = tile / ntn;
  const int ng = (tile % ntn) * 16;
  v8f c = wmma_gemm_tile(ws + O_ATTN + (size_t)(mt * 16) * NQ, NQ,
                         wo + (size_t)ng * NQ, NQ, NQ);
  wmma_store_tile(c, out + (size_t)(mt * 16) * D + ng, D);
}

// ---------------- host launcher --------------------------------------------
extern "C" void kernel_launch(void* const* d_in, const int* in_sizes, int n_in,
                              void* d_out, int out_size, void* d_ws,
                              size_t ws_size, hipStream_t stream) {
  const float* x      = (const float*)d_in[0];
  const float* cacheK = (const float*)d_in[1];
  const float* cacheV = (const float*)d_in[2];
  const float* wq     = (const float*)d_in[3];
  const float* wk     = (const float*)d_in[4];
  const float* wv     = (const float*)d_in[5];
  const float* wo     = (const float*)d_in[6];

  float* out  = (float*)d_out;                           // (B,S,D)
  float* outK = out + (size_t)B * S * D;                 // (B,KVH,TTOT,HD)
  float* outV = outK + (size_t)B * KVH * TTOT * HD;
  float* ws   = (float*)d_ws;

  qkv_proj_kernel<<<96, 256, 0, stream>>>(x, wq, wk, wv, ws);
  {
    const int total = M * NQ + 2 * M * NKV;              // 196608
    rope_scatter_kernel<<<(total + 255) / 256, 256, 0, stream>>>(ws, outK, outV);
  }
  flash_wmma_kernel<<<B * KVH * NCHUNK, 32, 0, stream>>>(cacheK, cacheV, ws,
                                                         outK, outV);
  combine_kernel<<<B * H * S, HD, 0, stream>>>(ws);
  out_proj_kernel<<<40, 256, 0, stream>>>(ws, wo, out);
}